// GraphConvolutionalNeuralNetwork_43757126811863
// MI455X (gfx1250) — compile-verified
//
#include <hip/hip_runtime.h>
#include <hip/hip_bf16.h>

typedef float v2f __attribute__((ext_vector_type(2)));
typedef float v8f __attribute__((ext_vector_type(8)));

#define N_NODES   100000
#define N_EDGES   1600000
#define HID       128
#define N_CLASSES 32
#define N_GRAPHS  1024

__device__ __forceinline__ void atomAddF(float* p, float v) {
    __hip_atomic_fetch_add(p, v, __ATOMIC_RELAXED, __HIP_MEMORY_SCOPE_AGENT);
}

// ---------------- fill ----------------
__global__ void fill_kernel(float* __restrict__ p, float v, int n) {
    int i = blockIdx.x * blockDim.x + threadIdx.x;
    if (i < n) p[i] = v;
}

// ---------------- degree (in-degree, self-loop pre-seeded with 1.0) --------
__global__ void degree_kernel(const long long* __restrict__ dst,
                              float* __restrict__ deg, int e) {
    int i = blockIdx.x * blockDim.x + threadIdx.x;
    if (i < e) atomAddF(&deg[(int)dst[i]], 1.0f);
}

// ---------------- deg -> rsqrt(deg), in place ----------------
__global__ void rsqrt_kernel(float* __restrict__ d, int n) {
    int i = blockIdx.x * blockDim.x + threadIdx.x;
    if (i < n) d[i] = rsqrtf(d[i]);   // deg >= 1 always (self loop)
}

// ---------------- fp32 WMMA GEMM: C[M,N] = A[M,K] @ W[K,N] (+bias) ---------
// One wave32 computes a 16x16 tile, K-loop in steps of 4 via
// v_wmma_f32_16x16x4_f32.  M,N multiples of 16; all lanes active.
__global__ void gemm_wmma_f32(const float* __restrict__ A,
                              const float* __restrict__ W,
                              float* __restrict__ C,
                              const float* __restrict__ bias,
                              int mtiles, int ntiles, int K, int N) {
    int wavesPerBlock = blockDim.x >> 5;
    int tile = blockIdx.x * wavesPerBlock + (threadIdx.x >> 5);
    if (tile >= mtiles * ntiles) return;          // wave-uniform: EXEC stays full
    int tr = tile / ntiles;
    int tc = tile % ntiles;
    int lane = threadIdx.x & 31;
    int half = lane >> 4;                          // 0: K pair {0,1}, 1: {2,3}
    int li   = lane & 15;
    int row = tr * 16 + li;                        // A-fragment row (M)
    int col = tc * 16 + li;                        // B-fragment col (N)

    v8f acc = {};
    for (int k0 = 0; k0 < K; k0 += 4) {
        int kk = k0 + half * 2;
        // A 16x4 fragment: lane holds A[row][kk], A[row][kk+1] (contiguous -> b64 load)
        v2f a = *(const v2f*)(A + (long)row * K + kk);
        // B 4x16 fragment: lane holds W[kk][col], W[kk+1][col]
        v2f b;
        b[0] = W[(long)kk * N + col];
        b[1] = W[(long)(kk + 1) * N + col];
        acc = __builtin_amdgcn_wmma_f32_16x16x4_f32(
            /*neg_a=*/false, a, /*neg_b=*/false, b,
            /*c_mod=*/(short)0, acc, /*reuse_a=*/false, /*reuse_b=*/false);
    }

    // C/D layout: VGPR i -> M = tr*16 + i + half*8, N = col
    int m0 = tr * 16 + half * 8;
    float bv = bias ? bias[col] : 0.0f;
#pragma unroll
    for (int i = 0; i < 8; ++i)
        C[(long)(m0 + i) * N + col] = acc[i] + bv;
}

// ---------------- edge scatter-add: out[dst] += h[src]*dinv[src]*dinv[dst] --
// One wave per edge; 32 lanes x float4 = 128 features.
__global__ void edge_agg_kernel(const float* __restrict__ h,
                                const long long* __restrict__ src,
                                const long long* __restrict__ dst,
                                const float* __restrict__ dinv,
                                float* __restrict__ out, int e) {
    int wid  = blockIdx.x * (blockDim.x >> 5) + (threadIdx.x >> 5);
    int lane = threadIdx.x & 31;
    if (wid >= e) return;
    int s = (int)src[wid];
    int d = (int)dst[wid];
    float norm = dinv[s] * dinv[d];
    const float4 hv = *(const float4*)(h + (long)s * HID + lane * 4);
    float* o = out + (long)d * HID + lane * 4;
    atomAddF(o + 0, hv.x * norm);
    atomAddF(o + 1, hv.y * norm);
    atomAddF(o + 2, hv.z * norm);
    atomAddF(o + 3, hv.w * norm);
}

// ---------------- self loop + bias (+optional relu) ----------------
__global__ void self_bias_kernel(float* __restrict__ out,
                                 const float* __restrict__ h,
                                 const float* __restrict__ dinv,
                                 const float* __restrict__ bias,
                                 int total, int do_relu) {
    int i = blockIdx.x * blockDim.x + threadIdx.x;
    if (i >= total) return;
    int v = i >> 7;          // /HID
    int f = i & (HID - 1);
    float di = dinv[v];
    float val = out[i] + h[i] * di * di + bias[f];
    out[i] = do_relu ? fmaxf(val, 0.0f) : val;
}

// ---------------- pooled sums + counts (one wave per node) -----------------
__global__ void pool_accum_kernel(const float* __restrict__ h,
                                  const long long* __restrict__ batch,
                                  float* __restrict__ pooled,
                                  float* __restrict__ counts, int n) {
    int node = blockIdx.x * (blockDim.x >> 5) + (threadIdx.x >> 5);
    int lane = threadIdx.x & 31;
    if (node >= n) return;
    int g = (int)batch[node];
    const float4 v = *(const float4*)(h + (long)node * HID + lane * 4);
    float* p = pooled + (long)g * HID + lane * 4;
    atomAddF(p + 0, v.x);
    atomAddF(p + 1, v.y);
    atomAddF(p + 2, v.z);
    atomAddF(p + 3, v.w);
    if (lane == 0) atomAddF(&counts[g], 1.0f);
}

__global__ void pool_div_kernel(float* __restrict__ pooled,
                                const float* __restrict__ counts, int total) {
    int i = blockIdx.x * blockDim.x + threadIdx.x;
    if (i >= total) return;
    int g = i >> 7;
    pooled[i] /= fmaxf(counts[g], 1.0f);
}

extern "C" void kernel_launch(void* const* d_in, const int* in_sizes, int n_in,
                              void* d_out, int out_size, void* d_ws, size_t ws_size,
                              hipStream_t stream) {
    const float*     x    = (const float*)d_in[0];
    const long long* eidx = (const long long*)d_in[1];   // int64 in reference
    const long long* bat  = (const long long*)d_in[2];   // int64 in reference
    const float*     W1   = (const float*)d_in[3];
    const float*     b1   = (const float*)d_in[4];
    const float*     W2   = (const float*)d_in[5];
    const float*     b2   = (const float*)d_in[6];
    const float*     Wf   = (const float*)d_in[7];
    const float*     bf   = (const float*)d_in[8];
    float* out = (float*)d_out;

    const long long* src = eidx;
    const long long* dst = eidx + N_EDGES;

    // workspace layout (floats)
    float* ws     = (float*)d_ws;
    float* dinv   = ws;                          // N_NODES (deg -> rsqrt in place)
    float* bufA   = dinv + N_NODES;              // N_NODES*HID
    float* bufB   = bufA + (size_t)N_NODES * HID;// N_NODES*HID
    float* pooled = bufB + (size_t)N_NODES * HID;// N_GRAPHS*HID
    float* counts = pooled + (size_t)N_GRAPHS * HID; // N_GRAPHS

    const int T = 256;
    const int NH = N_NODES * HID;
    const int mtiles = N_NODES / 16;             // 6250
    const int gemm_blocks = (mtiles * (HID / 16) + 7) / 8;  // 8 waves/block
    const int edge_blocks = (N_EDGES + 7) / 8;   // 1 wave per edge, 8 waves/block
    const int node_blocks = (N_NODES + 7) / 8;

    // ---- init (harness poisons d_ws; re-init every call) ----
    fill_kernel<<<(N_NODES + T - 1) / T, T, 0, stream>>>(dinv, 1.0f, N_NODES); // self loop
    fill_kernel<<<(NH + T - 1) / T, T, 0, stream>>>(bufB, 0.0f, NH);
    fill_kernel<<<(N_GRAPHS * HID + T - 1) / T, T, 0, stream>>>(pooled, 0.0f, N_GRAPHS * HID);
    fill_kernel<<<(N_GRAPHS + T - 1) / T, T, 0, stream>>>(counts, 0.0f, N_GRAPHS);

    // ---- symmetric normalization ----
    degree_kernel<<<(N_EDGES + T - 1) / T, T, 0, stream>>>(dst, dinv, N_EDGES);
    rsqrt_kernel<<<(N_NODES + T - 1) / T, T, 0, stream>>>(dinv, N_NODES);

    // ---- layer 1: h1 = x @ W1 ; aggregate ; +b1 ; relu ----
    gemm_wmma_f32<<<gemm_blocks, T, 0, stream>>>(x, W1, bufA, nullptr,
                                                 mtiles, HID / 16, HID, HID);
    edge_agg_kernel<<<edge_blocks, T, 0, stream>>>(bufA, src, dst, dinv, bufB, N_EDGES);
    self_bias_kernel<<<(NH + T - 1) / T, T, 0, stream>>>(bufB, bufA, dinv, b1, NH, 1);

    // ---- layer 2: h2 = relu_out @ W2 ; aggregate ; +b2 ----
    gemm_wmma_f32<<<gemm_blocks, T, 0, stream>>>(bufB, W2, bufA, nullptr,
                                                 mtiles, HID / 16, HID, HID);
    fill_kernel<<<(NH + T - 1) / T, T, 0, stream>>>(bufB, 0.0f, NH);
    edge_agg_kernel<<<edge_blocks, T, 0, stream>>>(bufA, src, dst, dinv, bufB, N_EDGES);
    self_bias_kernel<<<(NH + T - 1) / T, T, 0, stream>>>(bufB, bufA, dinv, b2, NH, 0);

    // ---- global mean pool ----
    pool_accum_kernel<<<node_blocks, T, 0, stream>>>(bufB, bat, pooled, counts, N_NODES);
    pool_div_kernel<<<(N_GRAPHS * HID + T - 1) / T, T, 0, stream>>>(pooled, counts,
                                                                    N_GRAPHS * HID);

    // ---- classifier: out = pooled @ Wf + bf  (1024x128x32) ----
    const int f_mtiles = N_GRAPHS / 16;          // 64
    const int f_ntiles = N_CLASSES / 16;         // 2
    gemm_wmma_f32<<<(f_mtiles * f_ntiles + 7) / 8, T, 0, stream>>>(
        pooled, Wf, out, bf, f_mtiles, f_ntiles, HID, N_CLASSES);
}